// VPmatrixPointsDepth_15187004359122
// MI455X (gfx1250) — compile-verified
//
#include <hip/hip_runtime.h>
#include <stdint.h>

#define HH 512
#define WW 512
#define HWSZ (HH * WW)        // 262144 = 2^18
#define OW 517                // 512 + 5 after five grow-by-1 min pools
#define NCAP 16384            // power-of-2 >= 13860 fg pixels max

typedef __attribute__((ext_vector_type(2))) float v2f;
typedef __attribute__((ext_vector_type(8))) float v8f;

// Monotonic float -> u32 map: order(float) == order(u32)
__device__ __forceinline__ unsigned encode_f32(float z) {
    int b = __float_as_int(z);
    return (unsigned)(b ^ ((b >> 31) | 0x80000000));
}

// ---- CDNA5 async global->LDS DMA (ASYNCcnt-tracked), via ISA mnemonics ----
__device__ __forceinline__ unsigned lds_off_of(const void* p) {
    return (unsigned)(uintptr_t)(__attribute__((address_space(3))) const void*)p;
}
__device__ __forceinline__ void async_g2l_b32(unsigned lds, const void* g) {
    asm volatile("global_load_async_to_lds_b32 %0, %1, off"
                 :: "v"(lds), "v"((unsigned long long)(uintptr_t)g) : "memory");
}
__device__ __forceinline__ void async_g2l_b64(unsigned lds, const void* g) {
    asm volatile("global_load_async_to_lds_b64 %0, %1, off"
                 :: "v"(lds), "v"((unsigned long long)(uintptr_t)g) : "memory");
}
__device__ __forceinline__ void async_wait0() {
    asm volatile("s_wait_asynccnt 0x0" ::: "memory");
}

// ---------------------------------------------------------------- init
__global__ void k_init(unsigned* zbuf, unsigned* counts, long long nz, int B) {
    long long i = (long long)blockIdx.x * blockDim.x + threadIdx.x;
    if (i < nz) zbuf[i] = 0xFF800000u;   // encode(+inf)
    if (i < B)  counts[i] = 0u;
}

// ---------------------------------------------------------------- VP = P @ V
__global__ void k_vp(const float* __restrict__ P, const float* __restrict__ V,
                     float* __restrict__ VP, int B) {
    int id = blockIdx.x * blockDim.x + threadIdx.x;
    if (id >= B * 16) return;
    int b = id >> 4, i = (id >> 2) & 3, j = id & 3;
    const float* Pb = P + b * 16;
    const float* Vb = V + b * 16;
    float acc = 0.f;
#pragma unroll
    for (int k = 0; k < 4; ++k) acc += Pb[i * 4 + k] * Vb[k * 4 + j];
    VP[b * 16 + i * 4 + j] = acc;
}

// ------------------------------------------------- WMMA transform + scatter-min
// One wave: 16 points x 4 batches via V_WMMA_F32_16X16X4_F32.
// A (16x4): lanes 0-15 -> M=lane, {K0,K1}; lanes 16-31 -> {K2,K3}  (ISA layout)
// B (4x16): column n = (b_local<<2)|j carries VP[b][j][k] over K.
// D (16x16): lane n = lane&15 column; vgpr r -> M = r + 8*(lane>=16).
__global__ void __launch_bounds__(32)
k_transform(const float* __restrict__ pts, const float* __restrict__ VP,
            unsigned* __restrict__ zbuf, int n_points) {
    const int lane = threadIdx.x;
    const int tile = blockIdx.x;
    const int bg   = blockIdx.y;
    const int lh   = lane >> 4;      // lane half
    const int m0   = lane & 15;

    int pidx   = tile * 16 + m0;
    int pclamp = (pidx < n_points) ? pidx : (n_points - 1);
    v2f a;
    a[0] = pts[pclamp * 4 + (lh ? 2 : 0)];
    a[1] = pts[pclamp * 4 + (lh ? 3 : 1)];

    const int n  = lane & 15;
    const int b  = bg * 4 + (n >> 2);
    const int j  = n & 3;
    v2f bm;
    bm[0] = VP[b * 16 + j * 4 + (lh ? 2 : 0)];
    bm[1] = VP[b * 16 + j * 4 + (lh ? 3 : 1)];

    v8f c = {};
    v8f d = __builtin_amdgcn_wmma_f32_16x16x4_f32(
        /*neg_a=*/false, a, /*neg_b=*/false, bm,
        /*c_mod=*/(short)0, c, /*reuse_a=*/false, /*reuse_b=*/false);

    const int base = lane & ~3;      // 4-lane group: columns j=0..3 of batch b
    const int t    = lane & 3;
#pragma unroll
    for (int r = 0; r < 8; ++r) {
        float x = __shfl(d[r], base + 0, 32);
        float y = __shfl(d[r], base + 1, 32);
        float z = __shfl(d[r], base + 2, 32);
        float w = __shfl(d[r], base + 3, 32);
        if (t == (r & 3)) {          // distribute 8 points over 4 lanes
            int gp = tile * 16 + lh * 8 + r;
            if (gp < n_points) {
                bool  nzw = (w != 0.0f);
                float nx = nzw ? x / w : x;
                float ny = nzw ? y / w : y;
                float nz = nzw ? z / w : z;
                int sx = (int)rintf((nx + 1.0f) * 0.5f * (float)WW);
                int sy = (int)rintf((1.0f - (ny + 1.0f) * 0.5f) * (float)HH);
                if (sx >= 0 && sx < WW && sy >= 0 && sy < HH) {
                    atomicMin(&zbuf[(size_t)b * HWSZ + sy * WW + sx],
                              encode_f32(nz));
                }
            }
        }
    }
}

// ------------------------------------------- compact fg pixels; zbuf -> dm0(=1.0)
__global__ void k_compact(unsigned* __restrict__ zbuf, unsigned* __restrict__ counts,
                          unsigned long long* __restrict__ lists, long long nz) {
    long long i = (long long)blockIdx.x * blockDim.x + threadIdx.x;
    if (i >= nz) return;
    unsigned u   = zbuf[i];
    int      b   = (int)(i >> 18);          // HWSZ == 2^18
    unsigned pix = (unsigned)(i & (HWSZ - 1));
    ((float*)zbuf)[i] = 1.0f;               // background value; fg overwritten later
    if (u < 0xC0000000u) {                  // encode(2.0f): z < 2.0 strict
        unsigned pos = atomicAdd(&counts[b], 1u);
        if (pos < NCAP)
            lists[(size_t)b * NCAP + pos] =
                ((unsigned long long)u << 32) | (unsigned long long)pix;
    }
}

// --------------------------------------- per-batch LDS bitonic sort + rank write
// Key list staged into 128KB LDS with GLOBAL_LOAD_ASYNC_TO_LDS_B64 (ASYNCcnt).
__global__ void __launch_bounds__(1024)
k_sortrank(const unsigned long long* __restrict__ lists,
           const unsigned* __restrict__ counts, float* __restrict__ dm0) {
    extern __shared__ unsigned long long s[];   // NCAP u64 = 128 KB
    const int b   = blockIdx.x;
    const int tid = threadIdx.x;
    unsigned K = counts[b];
    if (K > NCAP) K = NCAP;

    for (int i = tid; i < NCAP; i += 1024) {
        if (i < (int)K)
            async_g2l_b64(lds_off_of(&s[i]), &lists[(size_t)b * NCAP + i]);
        else
            s[i] = 0xFFFFFFFFFFFFFFFFull;       // sort-to-end padding
    }
    async_wait0();
    __syncthreads();

    for (unsigned k = 2; k <= NCAP; k <<= 1) {
        for (unsigned jj = k >> 1; jj > 0; jj >>= 1) {
            for (unsigned i = tid; i < NCAP; i += 1024) {
                unsigned ixj = i ^ jj;
                if (ixj > i) {
                    unsigned long long av = s[i], bv = s[ixj];
                    bool up = ((i & k) == 0);
                    if ((av > bv) == up) { s[i] = bv; s[ixj] = av; }
                }
            }
            __syncthreads();
        }
    }

    float denom = (K > 1) ? (float)(K - 1) : 1.0f;
    for (unsigned r = tid; r < K; r += 1024) {
        unsigned pix = (unsigned)(s[r] & 0xFFFFFFFFull);
        float scaled = (K > 1) ? ((1.0f - (float)r / denom) * 0.6f + 0.2f) : 0.5f;
        dm0[(size_t)b * HWSZ + pix] = scaled;
    }
}

// -------- fused 5x(4x4 min-pool, pad=1.0)  ==  separable 16-tap min, window [-10,+5]
// Input tile staged into LDS with GLOBAL_LOAD_ASYNC_TO_LDS_B32.
__global__ void __launch_bounds__(256)
k_erode(const float* __restrict__ dm0, float* __restrict__ out) {
    __shared__ float tile[31][32];
    __shared__ float rm[31][16];
    const int b  = blockIdx.z;
    const int ox = blockIdx.x * 16, oy = blockIdx.y * 16;
    const int tx = threadIdx.x, ty = threadIdx.y;
    const int t  = ty * 16 + tx;
    const float* in = dm0 + (size_t)b * HWSZ;

    // gfx1250 prefetch of the tile's input region (global_prefetch_b8)
    {
        int py = oy - 10 + ty * 2;
        int px = ox - 10 + tx * 2;
        if (py >= 0 && py < HH && px >= 0 && px < WW)
            __builtin_prefetch(&in[py * WW + px], 0, 0);
    }

    for (int idx = t; idx < 31 * 31; idx += 256) {
        int r = idx / 31, cc = idx % 31;
        int iy = oy - 10 + r, ix = ox - 10 + cc;
        if (iy >= 0 && iy < HH && ix >= 0 && ix < WW)
            async_g2l_b32(lds_off_of(&tile[r][cc]), &in[iy * WW + ix]);
        else
            tile[r][cc] = 1.0f;                 // halo / OOB value
    }
    async_wait0();
    __syncthreads();

    for (int idx = t; idx < 31 * 16; idx += 256) {
        int r = idx >> 4, cc = idx & 15;
        float m = tile[r][cc];
#pragma unroll
        for (int d2 = 1; d2 < 16; ++d2) m = fminf(m, tile[r][cc + d2]);
        rm[r][cc] = m;
    }
    __syncthreads();

    float m = rm[ty][tx];
#pragma unroll
    for (int d2 = 1; d2 < 16; ++d2) m = fminf(m, rm[ty + d2][tx]);
    int Y = oy + ty, X = ox + tx;
    if (Y < OW && X < OW)
        out[(size_t)b * OW * OW + (size_t)Y * OW + X] = m;
}

// ----------------------------------------------------------------- launcher
extern "C" void kernel_launch(void* const* d_in, const int* in_sizes, int n_in,
                              void* d_out, int out_size, void* d_ws, size_t ws_size,
                              hipStream_t stream) {
    const float* V   = (const float*)d_in[0];   // (B,4,4)
    const float* P   = (const float*)d_in[1];   // (B,4,4)
    const float* pts = (const float*)d_in[2];   // (N,4)
    const int B        = in_sizes[0] / 16;      // 128
    const int n_points = in_sizes[2] / 4;       // 13860
    float* out = (float*)d_out;                 // (B,1,517,517)

    // workspace layout
    char* ws = (char*)d_ws;
    size_t zbytes = (size_t)B * HWSZ * 4;                       // 128 MB
    unsigned*            zbuf   = (unsigned*)ws;                // also dm0 later
    unsigned*            counts = (unsigned*)(ws + zbytes);
    float*               VPb    = (float*)(ws + zbytes + 512);
    unsigned long long*  lists  = (unsigned long long*)(ws + zbytes + 512 + 8192);

    const long long nz = (long long)B * HWSZ;
    const unsigned gInit = (unsigned)((nz + 255) / 256);

    k_init<<<gInit, 256, 0, stream>>>(zbuf, counts, nz, B);
    k_vp<<<(B * 16 + 255) / 256, 256, 0, stream>>>(P, V, VPb, B);

    dim3 gT((n_points + 15) / 16, B / 4);
    k_transform<<<gT, 32, 0, stream>>>(pts, VPb, zbuf, n_points);

    k_compact<<<gInit, 256, 0, stream>>>(zbuf, counts, lists, nz);

    k_sortrank<<<B, 1024, NCAP * sizeof(unsigned long long), stream>>>(
        lists, counts, (float*)zbuf);

    dim3 gE((OW + 15) / 16, (OW + 15) / 16, B);
    k_erode<<<gE, dim3(16, 16), 0, stream>>>((const float*)zbuf, out);
}